// YoloLoss_86595130622372
// MI455X (gfx1250) — compile-verified
//
#include <hip/hip_runtime.h>
#include <stdint.h>

// Problem constants from the reference: B=64, N=25200, D=21, G=32, C=16
#define Bc 64
#define Nc 25200
#define Dc 21
#define Gc 32
#define CIOU_EPS 1e-6f
#define FOUR_OVER_PI2 0.405284734569351f
#define ITERS 8          // n's per thread
#define CHUNK (256 * ITERS)

__device__ __forceinline__ float frcp(float x) {
    // v_rcp_f32: TRANS unit, co-executes with VALU — much cheaper than IEEE div.
    return __builtin_amdgcn_rcpf(x);
}

// Map float to uint so that unsigned compare == float compare (total order).
__device__ __forceinline__ unsigned int fkey(float f) {
    unsigned int u = __float_as_uint(f);
    return (u & 0x80000000u) ? ~u : (u | 0x80000000u);
}

__device__ __forceinline__ float softplus_neg(float x) {
    // softplus(-x) = max(-x,0) + log1p(exp(-|x|))  (numerically stable)
    return fmaxf(-x, 0.0f) + log1pf(expf(-fabsf(x)));
}

// ---------------------------------------------------------------------------
// Phase 1: for each (b,g) find argmax_n CIoU(pred[b,n], gt[b,g])
// grid = (ceil(N/CHUNK), B), block = 256 (8 wave32)
// Register-tiled: 8 predictions decoded into VGPRs once, g is the OUTER loop,
// per-g result reduced immediately (no 32-entry accumulator array).
// ---------------------------------------------------------------------------
__global__ __launch_bounds__(256) void k_ciou_argmax(
        const float* __restrict__ preds,         // [B,N,21]
        const float* __restrict__ gt_boxes,      // [B,G,4]
        unsigned long long* __restrict__ best)   // [B,G] packed (key<<32)|n
{
    const int b   = blockIdx.y;
    const int tid = threadIdx.x;

    __shared__ float sraw[Gc * 4];
    __shared__ float sgx[Gc], sgy[Gc], sgx1[Gc], sgy1[Gc], sgx2[Gc], sgy2[Gc];
    __shared__ float sgar[Gc], sgat[Gc];
    __shared__ unsigned long long sred[Gc];

    if (tid < Gc) sred[tid] = 0ull;

    // --- Stage GT boxes (128 dwords) into LDS via gfx1250 async global->LDS DMA
    if (tid < Gc * 4) {
        const float* src = gt_boxes + ((size_t)b * Gc * 4 + tid);
        unsigned int       lds_off = (unsigned int)(uintptr_t)(&sraw[tid]);
        unsigned long long gaddr   = (unsigned long long)(uintptr_t)src;
        asm volatile("global_load_async_to_lds_b32 %0, %1, off"
                     :: "v"(lds_off), "v"(gaddr) : "memory");
    }
    asm volatile("s_wait_asynccnt 0" ::: "memory");
    __syncthreads();

    // --- Per-g precompute (hoisted out of the 51.6M-pair inner loop)
    if (tid < Gc) {
        float x = sraw[tid * 4 + 0], y = sraw[tid * 4 + 1];
        float w = sraw[tid * 4 + 2], h = sraw[tid * 4 + 3];
        sgx[tid]  = x;              sgy[tid]  = y;
        sgx1[tid] = x - 0.5f * w;   sgy1[tid] = y - 0.5f * h;
        sgx2[tid] = x + 0.5f * w;   sgy2[tid] = y + 0.5f * h;
        sgar[tid] = w * h;
        sgat[tid] = atanf(w * frcp(h));
    }
    __syncthreads();

    // --- Decode this thread's ITERS predictions into registers (once)
    const int base = blockIdx.x * CHUNK;
    float px[ITERS], py[ITERS], px1[ITERS], py1[ITERS], px2[ITERS], py2[ITERS];
    float par[ITERS], pat[ITERS];
    int   pn[ITERS];
    bool  pv[ITERS];
    #pragma unroll
    for (int i = 0; i < ITERS; ++i) {
        int n  = base + i * 256 + tid;
        pv[i]  = (n < Nc);
        int nc = pv[i] ? n : (Nc - 1);     // clamp load address for tail lanes
        pn[i]  = n;
        const float* p = preds + ((size_t)b * Nc + nc) * Dc;
        float x = p[0], y = p[1], w = p[2], h = p[3];
        px[i]  = x;             py[i]  = y;
        px1[i] = x - 0.5f * w;  py1[i] = y - 0.5f * h;
        px2[i] = x + 0.5f * w;  py2[i] = y + 0.5f * h;
        par[i] = w * h;
        pat[i] = atanf(w * frcp(h));
    }

    // --- g is the outer loop: 8 uniform LDS reads per g, then 8 in-register
    //     CIoU evaluations, then an immediate wave32 reduction for that g.
    #pragma unroll 1
    for (int g = 0; g < Gc; ++g) {
        const float gx = sgx[g],  gy = sgy[g];
        const float gx1 = sgx1[g], gy1 = sgy1[g];
        const float gx2 = sgx2[g], gy2 = sgy2[g];
        const float gareps = sgar[g] + CIOU_EPS;
        const float gat = sgat[g];

        float bv = -__builtin_inff();
        int   bn = 0;
        #pragma unroll
        for (int i = 0; i < ITERS; ++i) {
            float iw = fmaxf(fminf(px2[i], gx2) - fmaxf(px1[i], gx1), 0.0f);
            float ih = fmaxf(fminf(py2[i], gy2) - fmaxf(py1[i], gy1), 0.0f);
            float inter = iw * ih;
            float uni   = par[i] + gareps - inter;
            float iou   = inter * frcp(uni);
            float dx = px[i] - gx, dy = py[i] - gy;
            float cd = dx * dx + dy * dy;
            float ew = fmaxf(px2[i], gx2) - fminf(px1[i], gx1);
            float eh = fmaxf(py2[i], gy2) - fminf(py1[i], gy1);
            float dg = ew * ew + eh * eh + CIOU_EPS;
            float dv = pat[i] - gat;
            float v  = FOUR_OVER_PI2 * dv * dv;
            float alpha = v * frcp(1.0f - iou + v + CIOU_EPS);
            float c = iou - cd * frcp(dg) - alpha * v;
            c = pv[i] ? c : -__builtin_inff();   // neutralize tail lanes
            if (c > bv) { bv = c; bn = pn[i]; }
        }

        // wave32 shuffle reduction of (bv, bn)
        for (int off = 16; off > 0; off >>= 1) {
            float ov = __shfl_down(bv, off, 32);
            int   on = __shfl_down(bn, off, 32);
            if (ov > bv) { bv = ov; bn = on; }
        }
        if ((tid & 31) == 0) {
            unsigned long long pk =
                ((unsigned long long)fkey(bv) << 32) | (unsigned int)bn;
            atomicMax(&sred[g], pk);             // ds_max_u64
        }
    }

    __syncthreads();
    if (tid < Gc) atomicMax(&best[(size_t)b * Gc + tid], sred[tid]);
}

// ---------------------------------------------------------------------------
// Phase 2: gather matched predictions, compute the three loss terms, sum.
// grid = B, block = 32 (one wave per batch)
// ---------------------------------------------------------------------------
__global__ __launch_bounds__(32) void k_finalize(
        const float* __restrict__ preds,
        const float* __restrict__ gt_boxes,
        const int*   __restrict__ gt_cls,
        const unsigned long long* __restrict__ best,
        float* __restrict__ out)
{
    const int b = blockIdx.x;
    const int g = threadIdx.x;

    const unsigned long long pk = best[(size_t)b * Gc + g];
    const int n = (int)(pk & 0xFFFFFFFFull);

    const float* p  = preds    + ((size_t)b * Nc + n) * Dc;
    const float* gb = gt_boxes + ((size_t)b * Gc + g) * 4;

    const float x = p[0], y = p[1], w = p[2], h = p[3];
    const float gx = gb[0], gy = gb[1], gw = gb[2], gh = gb[3];

    const float px1 = x - 0.5f * w,   py1 = y - 0.5f * h;
    const float px2 = x + 0.5f * w,   py2 = y + 0.5f * h;
    const float gx1 = gx - 0.5f * gw, gy1 = gy - 0.5f * gh;
    const float gx2 = gx + 0.5f * gw, gy2 = gy + 0.5f * gh;

    float iw = fmaxf(fminf(px2, gx2) - fmaxf(px1, gx1), 0.0f);
    float ih = fmaxf(fminf(py2, gy2) - fmaxf(py1, gy1), 0.0f);
    float inter = iw * ih;
    float uni   = w * h + gw * gh - inter + CIOU_EPS;
    float iou   = inter * frcp(uni);
    float dx = x - gx, dy = y - gy;
    float cd = dx * dx + dy * dy;
    float ew = fmaxf(px2, gx2) - fminf(px1, gx1);
    float eh = fmaxf(py2, gy2) - fminf(py1, gy1);
    float dg = ew * ew + eh * eh + CIOU_EPS;
    float dv = atanf(w * frcp(h)) - atanf(gw * frcp(gh));
    float v  = FOUR_OVER_PI2 * dv * dv;
    float alpha = v * frcp(1.0f - iou + v + CIOU_EPS);
    float ciou  = iou - cd * frcp(dg) - alpha * v;

    const int cls = gt_cls[b * Gc + g];
    float loss = 5.0f * (1.0f - ciou)          // lambda_box * loss_box
               + softplus_neg(p[4])            // lambda_obj * loss_obj
               + softplus_neg(p[5 + cls]);     // lambda_cls * loss_cls

    for (int off = 16; off > 0; off >>= 1)
        loss += __shfl_down(loss, off, 32);
    if (g == 0) atomicAdd(out, loss);
}

// ---------------------------------------------------------------------------
extern "C" void kernel_launch(void* const* d_in, const int* in_sizes, int n_in,
                              void* d_out, int out_size, void* d_ws, size_t ws_size,
                              hipStream_t stream) {
    const float* preds    = (const float*)d_in[0];
    const float* gt_boxes = (const float*)d_in[1];
    const int*   gt_cls   = (const int*)d_in[2];
    float* out = (float*)d_out;
    unsigned long long* best = (unsigned long long*)d_ws;

    (void)in_sizes; (void)n_in; (void)out_size; (void)ws_size;

    // Deterministic per-call init (graph-capture safe)
    hipMemsetAsync(best, 0, (size_t)Bc * Gc * sizeof(unsigned long long), stream);
    hipMemsetAsync(out, 0, sizeof(float), stream);

    dim3 grid((Nc + CHUNK - 1) / CHUNK, Bc);
    k_ciou_argmax<<<grid, 256, 0, stream>>>(preds, gt_boxes, best);
    k_finalize<<<Bc, 32, 0, stream>>>(preds, gt_boxes, gt_cls, best, out);
}